// ReflexiveConsciousnessV2_57166014709763
// MI455X (gfx1250) — compile-verified
//
#include <hip/hip_runtime.h>
#include <math.h>

typedef float v2f __attribute__((ext_vector_type(2)));
typedef float v8f __attribute__((ext_vector_type(8)));

#define S_STREAMS 8
#define NCLQ      255
#define NPAD      256
#define DDIM      32
#define BATCH     1024
#define NITER     5
#define LSTR      40          // LDS row stride (floats) for WMMA B-operand arrays
#define PI_F      3.14159265358979323846f

__device__ __forceinline__ float softplusf(float x) {
  return (x > 20.0f) ? x : log1pf(expf(x));
}

// D = A(16x4 f32) * B(4x16 f32) + C(16x16 f32), wave32 striped.
__device__ __forceinline__ v8f wmma_f32_16x16x4(v2f a, v2f b, v8f c) {
  return __builtin_amdgcn_wmma_f32_16x16x4_f32(false, a, false, b, (short)0, c,
                                               false, false);
}

// ---------------------------------------------------------------------------
// Kernel 0: build K (padded 256x256), krow, A0 = softplus(psi_A0), Phi0.
// K[i][j] = exp(-popcount((i+1)^(j+1)) / 8), zero pad row/col 255.
// ---------------------------------------------------------------------------
__global__ void k_setup(const float* __restrict__ psiA0,
                        const float* __restrict__ psiPhi0,
                        float* __restrict__ Kg, float* __restrict__ krow,
                        float* __restrict__ A0, float* __restrict__ Phi0) {
  const int i = threadIdx.x;  // 0..255
  float rs = 0.0f;
  for (int j = 0; j < NPAD; ++j) {
    float v = 0.0f;
    if (i < NCLQ && j < NCLQ) {
      int h = __popc((unsigned)((i + 1) ^ (j + 1)));
      v = expf(-(float)h * (1.0f / (float)S_STREAMS));
    }
    Kg[i * NPAD + j] = v;
    rs += v;
  }
  krow[i] = rs;
  for (int d = 0; d < DDIM; ++d) {
    float a0 = 0.0f, p0 = 0.0f;
    if (i < NCLQ) {
      a0 = softplusf(psiA0[i * DDIM + d]);
      p0 = psiPhi0[i * DDIM + d];
    }
    A0[i * DDIM + d] = a0;
    Phi0[i * DDIM + d] = p0;
  }
}

// ---------------------------------------------------------------------------
// Kernel 1: feats[b][n] = X[1024,512] @ W_streams^T[512,256] + b_streams
// One 16x16 output tile per wave, f32 WMMA, K-loop of 128 x (K=4).
// Grid: 128 blocks x 256 threads (8 waves) = 1024 tiles.
// ---------------------------------------------------------------------------
__global__ void __launch_bounds__(256) k_feats(const float* __restrict__ X,
                                               const float* __restrict__ Ws,
                                               const float* __restrict__ bs,
                                               float* __restrict__ feats) {
  const int lane = threadIdx.x & 31;
  const int wave = threadIdx.x >> 5;
  const int tile = blockIdx.x * 8 + wave;  // 0..1023
  const int tm = tile >> 4;                // 64 M-tiles
  const int tn = tile & 15;                // 16 N-tiles
  const int mrow = tm * 16 + (lane & 15);
  const int ncol = tn * 16 + (lane & 15);
  const int koff = (lane >> 4) << 1;

  v8f acc = {0.f, 0.f, 0.f, 0.f, 0.f, 0.f, 0.f, 0.f};
  for (int kk = 0; kk < 128; ++kk) {
    const int kb = kk * 4 + koff;
    v2f a = *reinterpret_cast<const v2f*>(X + mrow * 512 + kb);
    v2f b = *reinterpret_cast<const v2f*>(Ws + ncol * 512 + kb);
    acc = wmma_f32_16x16x4(a, b, acc);
  }
  const float bias = bs[ncol];
#pragma unroll
  for (int rv = 0; rv < 8; ++rv) {
    const int m = tm * 16 + ((lane >> 4) << 3) + rv;
    feats[m * 256 + ncol] = acc[rv] + bias;
  }
}

// ---------------------------------------------------------------------------
// Kernel 2: clique features + input projections.
// One thread per (b, n): cf = mean of member stream feats; then
// A_input = softplus(cf @ W_A^T + b_A), phi_input = pi*tanh(cf @ W_phi^T + b_phi)
// ---------------------------------------------------------------------------
__global__ void __launch_bounds__(256) k_inputs(const float* __restrict__ feats,
                                                const float* __restrict__ W_A,
                                                const float* __restrict__ b_A,
                                                const float* __restrict__ W_phi,
                                                const float* __restrict__ b_phi,
                                                float* __restrict__ Ain,
                                                float* __restrict__ Phin) {
  const int gid = blockIdx.x * blockDim.x + threadIdx.x;  // b*255 + n
  if (gid >= BATCH * NCLQ) return;
  const int b = gid / NCLQ;
  const int n = gid - b * NCLQ;
  const unsigned mask = (unsigned)(n + 1);
  const float invd = 1.0f / (float)__popc(mask);

  float cf[DDIM];
#pragma unroll
  for (int d = 0; d < DDIM; ++d) cf[d] = 0.0f;
#pragma unroll
  for (int s = 0; s < S_STREAMS; ++s) {
    if ((mask >> s) & 1u) {
      const float* fr = feats + b * 256 + s * DDIM;
#pragma unroll
      for (int d = 0; d < DDIM; ++d) cf[d] += fr[d];
    }
  }
#pragma unroll
  for (int d = 0; d < DDIM; ++d) cf[d] *= invd;

  const size_t base = (size_t)gid * DDIM;
  for (int d = 0; d < DDIM; ++d) {
    float aa = b_A[d];
    float pp = b_phi[d];
    const float* wa = W_A + d * DDIM;
    const float* wp = W_phi + d * DDIM;
#pragma unroll
    for (int e = 0; e < DDIM; ++e) {
      aa += wa[e] * cf[e];
      pp += wp[e] * cf[e];
    }
    Ain[base + d] = softplusf(aa);
    Phin[base + d] = PI_F * tanhf(pp);
  }
}

// ---------------------------------------------------------------------------
// Kernel 3: fused 5-iteration evolution + readout. One workgroup per batch b.
// State A/phi live in LDS; K is L2-resident; the five K@R einsums share one
// WMMA k-loop (one K-tile fragment feeds 10 WMMAs).
// ---------------------------------------------------------------------------
__global__ void __launch_bounds__(256) k_main(
    const float* __restrict__ Kg, const float* __restrict__ krow,
    const float* __restrict__ A0, const float* __restrict__ Phi0,
    const float* __restrict__ Ain, const float* __restrict__ Phin,
    const float* __restrict__ Wc, const float* __restrict__ bc,
    const float* __restrict__ Wr, const float* __restrict__ br,
    const float* __restrict__ p_alpha, const float* __restrict__ p_msq,
    const float* __restrict__ p_pw, const float* __restrict__ p_iw,
    const float* __restrict__ p_ev, const float* __restrict__ p_pc,
    float* __restrict__ out) {
  __shared__ float sA[NPAD * LSTR];    // amplitude field (row 255 stays 0)
  __shared__ float sSp[NPAD * LSTR];   // sin(phi)
  __shared__ float sCp[NPAD * LSTR];   // cos(phi)
  __shared__ float sPhi[NPAD * DDIM];  // phase field
  __shared__ float sFin[1024 + 64];    // readout scratch

  const int t = threadIdx.x;
  const int lane = t & 31;
  const int wave = t >> 5;
  const int b = blockIdx.x;

  const float alpha = p_alpha[0];
  const float msq = p_msq[0];
  const float pw = p_pw[0];
  const float iw = p_iw[0];
  const float dt = 1.0f / (1.0f + expf(-p_ev[0]));
  const float pcoup = p_pc[0];
  const float invN = 1.0f / (float)NCLQ;

  // Init state (thread t owns row t; row 255 is zero from k_setup).
  {
    const int n = t;
#pragma unroll
    for (int d = 0; d < DDIM; ++d) {
      sA[n * LSTR + d] = A0[n * DDIM + d];
      sPhi[n * DDIM + d] = Phi0[n * DDIM + d];
    }
  }
  __syncthreads();

  const float* AinB = Ain + (size_t)b * (NCLQ * DDIM);
  const float* PhinB = Phin + (size_t)b * (NCLQ * DDIM);

  for (int it = 0; it < NITER; ++it) {
    // ---- phase 1: sin/cos of current phase (row per thread) ----
    {
      const int n = t;
#pragma unroll
      for (int d = 0; d < DDIM; ++d) {
        const float ph = sPhi[n * DDIM + d];
        sSp[n * LSTR + d] = sinf(ph);
        sCp[n * LSTR + d] = cosf(ph);
      }
    }
    __syncthreads();

    // ---- phase 2: K @ [A | sinφ | cosφ | A·cosφ | A·sinφ] via f32 WMMA ----
    float rA[2][2][8];  // new A, held until all waves finish reading LDS
    float rP[2][2][8];  // new phi
    for (int mi = 0; mi < 2; ++mi) {
      const int mt = wave * 2 + mi;           // m-tile (clique rows)
      const int mrow = mt * 16 + (lane & 15); // A-operand row
      const int koff = (lane >> 4) << 1;
      v8f acc[2][5];
#pragma unroll
      for (int nt = 0; nt < 2; ++nt)
#pragma unroll
        for (int r = 0; r < 5; ++r)
          acc[nt][r] = (v8f){0.f, 0.f, 0.f, 0.f, 0.f, 0.f, 0.f, 0.f};

      for (int kk = 0; kk < 64; ++kk) {
        const int kb = kk * 4 + koff;
        const v2f af = *reinterpret_cast<const v2f*>(Kg + mrow * NPAD + kb);
#pragma unroll
        for (int nt = 0; nt < 2; ++nt) {
          const int d = nt * 16 + (lane & 15);
          const float a0 = sA[kb * LSTR + d];
          const float a1 = sA[(kb + 1) * LSTR + d];
          const float s0 = sSp[kb * LSTR + d];
          const float s1 = sSp[(kb + 1) * LSTR + d];
          const float c0 = sCp[kb * LSTR + d];
          const float c1 = sCp[(kb + 1) * LSTR + d];
          const v2f bA = {a0, a1};
          const v2f bS = {s0, s1};
          const v2f bC = {c0, c1};
          const v2f bAC = {a0 * c0, a1 * c1};
          const v2f bAS = {a0 * s0, a1 * s1};
          acc[nt][0] = wmma_f32_16x16x4(af, bA, acc[nt][0]);   // K@A
          acc[nt][1] = wmma_f32_16x16x4(af, bS, acc[nt][1]);   // K@sin
          acc[nt][2] = wmma_f32_16x16x4(af, bC, acc[nt][2]);   // K@cos
          acc[nt][3] = wmma_f32_16x16x4(af, bAC, acc[nt][3]);  // K@(A cos)
          acc[nt][4] = wmma_f32_16x16x4(af, bAS, acc[nt][4]);  // K@(A sin)
        }
      }

      // elementwise field update at this wave's own (m,d) positions
#pragma unroll
      for (int nt = 0; nt < 2; ++nt) {
        const int d = nt * 16 + (lane & 15);
#pragma unroll
        for (int rv = 0; rv < 8; ++rv) {
          const int m = mt * 16 + ((lane >> 4) << 3) + rv;
          float An = 0.0f, Pn = 0.0f;
          if (m < NCLQ) {
            const float Aold = sA[m * LSTR + d];
            const float phio = sPhi[m * DDIM + d];
            const float spi = sSp[m * LSTR + d];
            const float cpi = sCp[m * LSTR + d];
            const float kr = krow[m];
            const float ain = AinB[m * DDIM + d];
            const float pin = PhinB[m * DDIM + d];
            const float lap = (acc[nt][0][rv] - kr * Aold) * invN;
            const float pc =
                Aold * (cpi * acc[nt][1][rv] - spi * acc[nt][2][rv]) * invN;
            const float V =
                (cpi * acc[nt][3][rv] + spi * acc[nt][4][rv]) * invN;
            const float drive = ain * sinf(pin - phio);
            const float dA =
                alpha * lap + pw * V - msq * Aold + iw * (ain - Aold);
            An = softplusf(Aold + dt * dA);
            const float dphi = alpha * pc / (An + 1e-8f) + iw * drive;
            Pn = phio + dt * dphi;
          }
          rA[mi][nt][rv] = An;
          rP[mi][nt][rv] = Pn;
        }
      }
    }
    __syncthreads();  // everyone done reading old sA/sSp/sCp

    // ---- write back updated state ----
#pragma unroll
    for (int mi = 0; mi < 2; ++mi) {
      const int mt = wave * 2 + mi;
#pragma unroll
      for (int nt = 0; nt < 2; ++nt) {
        const int d = nt * 16 + (lane & 15);
#pragma unroll
        for (int rv = 0; rv < 8; ++rv) {
          const int m = mt * 16 + ((lane >> 4) << 3) + rv;
          if (m < NCLQ) {
            sA[m * LSTR + d] = rA[mi][nt][rv];
            sPhi[m * DDIM + d] = rP[mi][nt][rv];
          }
        }
      }
    }
    __syncthreads();

    // ---- per-clique norm clip (row per thread) ----
    if (t < NCLQ) {
      float ss = 0.0f;
#pragma unroll
      for (int d = 0; d < DDIM; ++d) {
        const float v = sA[t * LSTR + d];
        ss += v * v;
      }
      float nr = fmaxf(sqrtf(ss), 1e-8f);
      const float sc = (nr > 1.0f) ? (1.0f / nr) : 1.0f;
#pragma unroll
      for (int d = 0; d < DDIM; ++d) sA[t * LSTR + d] *= sc;
    }
    __syncthreads();
  }

  // ---- readout: Ac[d][c] = relu(sum_n A[n][d] * Wc[c][n] + bc[c]) ----
  if (wave < 4) {
    const int mt = wave >> 1;  // d-tile
    const int nt = wave & 1;   // c-tile
    const int didx = mt * 16 + (lane & 15);
    const int cidx = nt * 16 + (lane & 15);
    const int koff = (lane >> 4) << 1;
    v8f acc = {0.f, 0.f, 0.f, 0.f, 0.f, 0.f, 0.f, 0.f};
    for (int kk = 0; kk < 64; ++kk) {
      const int kb = kk * 4 + koff;
      const v2f af = {sA[kb * LSTR + didx], sA[(kb + 1) * LSTR + didx]};
      const float b0 = (kb < NCLQ) ? Wc[cidx * NCLQ + kb] : 0.0f;
      const float b1 = (kb + 1 < NCLQ) ? Wc[cidx * NCLQ + kb + 1] : 0.0f;
      const v2f bf = {b0, b1};
      acc = wmma_f32_16x16x4(af, bf, acc);
    }
    const float bias = bc[cidx];
#pragma unroll
    for (int rv = 0; rv < 8; ++rv) {
      const int dd = mt * 16 + ((lane >> 4) << 3) + rv;
      sFin[dd * DDIM + cidx] = fmaxf(acc[rv] + bias, 0.0f);
    }
  }
  __syncthreads();

  // ro = flatten(Ac) @ W_read^T + b_read  (64 outputs, one thread each)
  if (t < 64) {
    float a = br[t];
    const float* wr = Wr + t * 1024;
    for (int i = 0; i < 1024; ++i) a += sFin[i] * wr[i];
    sFin[1024 + t] = a;
  }
  __syncthreads();

  if (t < DDIM) {
    const float ro0 = sFin[1024 + t];
    const float ro1 = sFin[1024 + 32 + t];
    const float amp = softplusf(ro0);
    const float ph = tanhf(ro1 * pcoup) * PI_F;
    // psi (complex64 interleaved), then amplitude, then phase
    out[(size_t)b * 64 + 2 * t] = amp * cosf(ph);
    out[(size_t)b * 64 + 2 * t + 1] = amp * sinf(ph);
    out[(size_t)BATCH * 64 + (size_t)b * DDIM + t] = amp;
    out[(size_t)BATCH * 96 + (size_t)b * DDIM + t] = ph;
  }
}

// ---------------------------------------------------------------------------
extern "C" void kernel_launch(void* const* d_in, const int* in_sizes, int n_in,
                              void* d_out, int out_size, void* d_ws,
                              size_t ws_size, hipStream_t stream) {
  (void)in_sizes; (void)n_in; (void)out_size; (void)ws_size;
  const float* X = (const float*)d_in[0];
  const float* Ws = (const float*)d_in[1];
  const float* bs = (const float*)d_in[2];
  const float* W_A = (const float*)d_in[3];
  const float* b_A = (const float*)d_in[4];
  const float* W_phi = (const float*)d_in[5];
  const float* b_phi = (const float*)d_in[6];
  const float* psiA0 = (const float*)d_in[7];
  const float* psiPhi0 = (const float*)d_in[8];
  const float* Wc = (const float*)d_in[9];
  const float* bc = (const float*)d_in[10];
  const float* Wr = (const float*)d_in[11];
  const float* br = (const float*)d_in[12];
  const float* alpha = (const float*)d_in[13];
  const float* msq = (const float*)d_in[14];
  const float* pw = (const float*)d_in[15];
  const float* iw = (const float*)d_in[16];
  const float* ev = (const float*)d_in[17];
  const float* pc = (const float*)d_in[18];

  float* ws = (float*)d_ws;
  float* Kg = ws;                    // 256*256
  float* krow = Kg + 65536;          // 256
  float* A0 = krow + 256;            // 256*32
  float* Phi0 = A0 + 8192;           // 256*32
  float* feats = Phi0 + 8192;        // 1024*256
  float* Ain = feats + 262144;       // 1024*255*32
  float* Phin = Ain + 8355840;       // 1024*255*32

  k_setup<<<1, 256, 0, stream>>>(psiA0, psiPhi0, Kg, krow, A0, Phi0);
  k_feats<<<128, 256, 0, stream>>>(X, Ws, bs, feats);
  k_inputs<<<1020, 256, 0, stream>>>(feats, W_A, b_A, W_phi, b_phi, Ain, Phin);
  k_main<<<BATCH, 256, 0, stream>>>(Kg, krow, A0, Phi0, Ain, Phin, Wc, bc, Wr,
                                    br, alpha, msq, pw, iw, ev, pc,
                                    (float*)d_out);
}